// customGRUCell_23210003268047
// MI455X (gfx1250) — compile-verified
//
#include <hip/hip_runtime.h>
#include <math.h>

// Problem constants (from reference)
#define HH   1024
#define BB   64
#define II   128
#define BETA_C   (10.0f / 0.192f)
#define DT_C     0.1f
#define ZMIN_C   0.001f
#define ZRNG_C   0.099f   // Z_MAX - Z_MIN

typedef float v2f __attribute__((ext_vector_type(2)));
typedef float v8f __attribute__((ext_vector_type(8)));

__device__ __forceinline__ float sigm(float v) { return 1.0f / (1.0f + expf(-v)); }

// ---------------------------------------------------------------------------
// Kernel 1: WR[j,k] = log(cosh(BETA*raw))/BETA  (stable form)
// ---------------------------------------------------------------------------
__global__ void k_logcosh(const float* __restrict__ raw, float* __restrict__ WR) {
    int i = blockIdx.x * blockDim.x + threadIdx.x;   // exact grid: H*H
    float t  = BETA_C * raw[i];
    float ax = fabsf(t);
    WR[i] = (ax + log1pf(expf(-2.0f * ax)) - 0.6931471805599453f) * (1.0f / BETA_C);
}

// ---------------------------------------------------------------------------
// Kernel 2: rT[b,j] = sigmoid(v_t[b,j]);  S[j,b] = A[j]*r[j,b]
// ---------------------------------------------------------------------------
__global__ void k_rates(const float* __restrict__ v_t,
                        const float* __restrict__ a_exc,
                        const float* __restrict__ a_inh,
                        float* __restrict__ rT, float* __restrict__ S) {
    int idx = blockIdx.x * blockDim.x + threadIdx.x;  // H*B
    int b = idx / HH, j = idx % HH;
    float rv = sigm(v_t[idx]);
    rT[idx] = rv;
    float a = (j < HH / 2) ? a_exc[0] : a_inh[0];
    float A = 10.0f * sigm(a);
    S[j * BB + b] = A * rv;
}

// ---------------------------------------------------------------------------
// Kernel 3: WMMA f32 16x16x4 GEMMs.
//   acc1 = w_z @ S + |p_r| @ x  -> ZT = DT*sigmoid(acc1 + g_z)
//   acc2 = p_r @ x              -> PQ = acc2 + b_r
// One wave per 16x16 tile; block = 4 waves covering N=64, grid = H/16 blocks.
// A layout: M = lane&15, K = vgpr + 2*(lane>>4). B: N = lane&15, K = vgpr + 2*(lane>>4).
// C/D layout: M = vgpr + 8*(lane>>4), N = lane&15.
// ---------------------------------------------------------------------------
__global__ void k_gemm(const float* __restrict__ WR, const float* __restrict__ S,
                       const float* __restrict__ p_r, const float* __restrict__ x,
                       const float* __restrict__ g_z, const float* __restrict__ b_r,
                       float* __restrict__ ZT, float* __restrict__ PQ) {
    const int lane  = threadIdx.x & 31;
    const int wave  = threadIdx.x >> 5;        // 0..3 -> N tile
    const int j0    = blockIdx.x * 16;
    const int n0    = wave * 16;
    const int laneM = lane & 15;
    const int laneH = lane >> 4;               // 0 or 1
    const int rowA  = j0 + laneM;

    v8f acc1 = {};
    v8f acc2 = {};

    // Input-projection terms over K = I = 128
    for (int k0 = 0; k0 < II; k0 += 4) {
        int kA = k0 + 2 * laneH;
        float a0 = p_r[rowA * II + kA];
        float a1 = p_r[rowA * II + kA + 1];
        v2f a;  a[0] = a0;        a[1] = a1;
        v2f az; az[0] = fabsf(a0); az[1] = fabsf(a1);
        v2f bb; bb[0] = x[kA * BB + n0 + laneM];
                bb[1] = x[(kA + 1) * BB + n0 + laneM];
        acc2 = __builtin_amdgcn_wmma_f32_16x16x4_f32(false, a,  false, bb, (short)0, acc2, false, false);
        acc1 = __builtin_amdgcn_wmma_f32_16x16x4_f32(false, az, false, bb, (short)0, acc1, false, false);
    }

    // Recurrent gate term over K = H = 1024 (w_z == w_r >= 0)
    for (int k0 = 0; k0 < HH; k0 += 4) {
        int kA = k0 + 2 * laneH;
        v2f a;  a[0] = WR[rowA * HH + kA];
                a[1] = WR[rowA * HH + kA + 1];
        v2f bb; bb[0] = S[kA * BB + n0 + laneM];
                bb[1] = S[(kA + 1) * BB + n0 + laneM];
        acc1 = __builtin_amdgcn_wmma_f32_16x16x4_f32(false, a, false, bb, (short)0, acc1, false, false);
    }

#pragma unroll
    for (int rI = 0; rI < 8; ++rI) {
        int m = rI + 8 * laneH;
        int j = j0 + m;
        int b = n0 + laneM;
        ZT[j * BB + b] = DT_C * sigm(acc1[rI] + g_z[j]);
        PQ[j * BB + b] = acc2[rI] + b_r[j];
    }
}

// ---------------------------------------------------------------------------
// Kernel 4: streaming update (HBM-bound, 512 MB of X/U).
// One block per row j; per-(j,k) params computed once into LDS, reused for
// all 64 batches. Each of 8 waves handles 8 batches with float4 loads and a
// wave32 shuffle reduction.
// ---------------------------------------------------------------------------
__device__ __forceinline__ float syn_term(float xv, float uv, float wr,
                                          float zx, float zu, float uc,
                                          float rj, float rk) {
    float Xn = zx + (1.0f - zx) * xv - uv * xv * rj;            // DELTA_T = 1
    float Un = uc * zu + (1.0f - zu) * uv + uc * (1.0f - uv) * rj;
    Un = fminf(fmaxf(Un, uc), 1.0f);
    return Xn * Un * wr * rk;
}

__global__ void k_main(const float* __restrict__ X, const float* __restrict__ U,
                       const float* __restrict__ WR,
                       const float* __restrict__ c_x, const float* __restrict__ c_u,
                       const float* __restrict__ c_U,
                       const float* __restrict__ rT, const float* __restrict__ ZT,
                       const float* __restrict__ PQ, const float* __restrict__ v_t,
                       float* __restrict__ out) {
    const int j   = blockIdx.x;
    const int tid = threadIdx.x;

    __shared__ __align__(16) float lwr[HH];
    __shared__ __align__(16) float lzx[HH];
    __shared__ __align__(16) float lzu[HH];
    __shared__ __align__(16) float luc[HH];

    const size_t rowOff = (size_t)j * HH;
    for (int k = tid; k < HH; k += 256) {
        lwr[k] = WR[rowOff + k];
        lzx[k] = ZMIN_C + ZRNG_C * sigm(c_x[rowOff + k]);
        lzu[k] = ZMIN_C + ZRNG_C * sigm(c_u[rowOff + k]);
        luc[k] = 0.9f * sigm(c_U[rowOff + k]);
    }
    __syncthreads();

    const int lane = tid & 31;
    const int wave = tid >> 5;                 // 0..7 -> 8 batches each
    const float4* lwr4 = (const float4*)lwr;
    const float4* lzx4 = (const float4*)lzx;
    const float4* lzu4 = (const float4*)lzu;
    const float4* luc4 = (const float4*)luc;

    for (int bi = 0; bi < 8; ++bi) {
        const int b = wave * 8 + bi;
        const size_t base = (size_t)b * HH * HH + rowOff;
        const float4* X4 = (const float4*)(X + base);
        const float4* U4 = (const float4*)(U + base);
        const float4* r4 = (const float4*)(rT + (size_t)b * HH);
        const float rj = rT[(size_t)b * HH + j];

        if (bi < 7) {  // prefetch next batch's row (global_prefetch_b8)
            __builtin_prefetch(X + base + (size_t)HH * HH + lane * 4, 0, 1);
            __builtin_prefetch(U + base + (size_t)HH * HH + lane * 4, 0, 1);
        }

        float p = 0.0f;
#pragma unroll
        for (int c = 0; c < 8; ++c) {
            int i4 = c * 32 + lane;
            float4 xv = X4[i4], uv = U4[i4], rk = r4[i4];
            float4 wr = lwr4[i4], zx = lzx4[i4], zu = lzu4[i4], uc = luc4[i4];
            p += syn_term(xv.x, uv.x, wr.x, zx.x, zu.x, uc.x, rj, rk.x);
            p += syn_term(xv.y, uv.y, wr.y, zx.y, zu.y, uc.y, rj, rk.y);
            p += syn_term(xv.z, uv.z, wr.z, zx.z, zu.z, uc.z, rj, rk.z);
            p += syn_term(xv.w, uv.w, wr.w, zx.w, zu.w, uc.w, rj, rk.w);
        }
        // wave32 reduction
        for (int off = 16; off; off >>= 1) p += __shfl_xor(p, off, 32);

        if (lane == 0) {
            float zt = ZT[j * BB + b];
            float pq = PQ[j * BB + b];
            float vt = v_t[(size_t)b * HH + j];
            out[(size_t)b * HH + j] = (1.0f - zt) * vt + DT_C * (p + pq);
        }
    }
}

// ---------------------------------------------------------------------------
extern "C" void kernel_launch(void* const* d_in, const int* in_sizes, int n_in,
                              void* d_out, int out_size, void* d_ws, size_t ws_size,
                              hipStream_t stream) {
    const float* x_in  = (const float*)d_in[0];
    const float* v_t   = (const float*)d_in[1];
    const float* X     = (const float*)d_in[2];
    const float* U     = (const float*)d_in[3];
    const float* raw   = (const float*)d_in[4];
    const float* p_r   = (const float*)d_in[5];
    const float* b_r   = (const float*)d_in[6];
    const float* g_z   = (const float*)d_in[7];
    const float* c_x   = (const float*)d_in[8];
    const float* c_u   = (const float*)d_in[9];
    const float* c_U   = (const float*)d_in[10];
    const float* a_exc = (const float*)d_in[11];
    const float* a_inh = (const float*)d_in[12];
    float* out = (float*)d_out;

    // workspace layout (floats)
    float* ws = (float*)d_ws;
    float* WR = ws;                       // H*H     = 1048576
    float* rT = WR + (size_t)HH * HH;     // H*B     = 65536
    float* S  = rT + (size_t)HH * BB;     // H*B
    float* ZT = S  + (size_t)HH * BB;     // H*B
    float* PQ = ZT + (size_t)HH * BB;     // H*B

    k_logcosh<<<(HH * HH) / 256, 256, 0, stream>>>(raw, WR);
    k_rates  <<<(HH * BB) / 256, 256, 0, stream>>>(v_t, a_exc, a_inh, rT, S);
    k_gemm   <<<HH / 16, 128, 0, stream>>>(WR, S, p_r, x_in, g_z, b_r, ZT, PQ);
    k_main   <<<HH, 256, 0, stream>>>(X, U, WR, c_x, c_u, c_U, rT, ZT, PQ, v_t, out);
}